// FraudGraphSAGE_15118284882426
// MI455X (gfx1250) — compile-verified
//
#include <hip/hip_runtime.h>

typedef float v2f __attribute__((ext_vector_type(2)));
typedef float v8f __attribute__((ext_vector_type(8)));

#define N_NODES 50000
#define N_EDGES 800000
#define IN_CH   96
#define HID     128

// ---------------------------------------------------------------------------
// Utility: zero a float buffer (harness poisons d_ws with 0xAA each timing run
// is NOT re-poisoned, so we must clear accumulators ourselves every call).
// ---------------------------------------------------------------------------
__global__ void zero_f32(float* __restrict__ p, int n) {
  int i = blockIdx.x * blockDim.x + threadIdx.x;
  if (i < n) p[i] = 0.0f;
}

// ---------------------------------------------------------------------------
// In-degree count via float atomics (one thread per edge).
// ---------------------------------------------------------------------------
__global__ void degree_kernel(const int* __restrict__ dst,
                              float* __restrict__ cnt, int E) {
  int e = blockIdx.x * blockDim.x + threadIdx.x;
  if (e < E) atomicAdd(&cnt[dst[e]], 1.0f);
}

// cnt -> 1 / max(cnt, 1)   (PyG mean aggregation semantics)
__global__ void inv_kernel(float* __restrict__ cnt, int n) {
  int i = blockIdx.x * blockDim.x + threadIdx.x;
  if (i < n) cnt[i] = 1.0f / fmaxf(cnt[i], 1.0f);
}

// ---------------------------------------------------------------------------
// Scatter-add: one wave32 per edge, lanes stride the feature channels.
// Gather of feat[src] is coalesced (32 consecutive floats per wave), and the
// atomic adds into agg[dst] are coalesced too -> this is the HBM-bound part.
// ---------------------------------------------------------------------------
__global__ void scatter_kernel(const float* __restrict__ feat,
                               const int* __restrict__ src,
                               const int* __restrict__ dst,
                               float* __restrict__ agg, int E, int d) {
  int wave = (blockIdx.x * blockDim.x + threadIdx.x) >> 5;
  int lane = threadIdx.x & 31;
  if (wave >= E) return;                 // wave-uniform branch
  int s = src[wave];
  int t = dst[wave];
  const float* fp = feat + (size_t)s * d;
  float*       ap = agg  + (size_t)t * d;
  for (int c = lane; c < d; c += 32)
    atomicAdd(&ap[c], fp[c]);
}

// ---------------------------------------------------------------------------
// Fused SAGE layer GEMM on the matrix pipes:
//   hout = relu( (agg * inv) @ Wl  +  hin @ Wr  +  b )
// One wave computes one 16x16 output tile with V_WMMA_F32_16X16X4_F32,
// accumulating both products into the same f32 accumulator. 50000 is an
// exact multiple of 16 and dout in {128,64} is a multiple of 16, so every
// tile is full and EXEC stays all-ones (WMMA requirement).
//
// Layouts (ISA 7.12.2, wave32):
//   A 16x4 f32: lane<16 -> M=lane, K={k0,k0+1}; lane>=16 -> M=lane-16, K={k0+2,k0+3}
//   B 4x16 f32: lane<16 -> N=lane, K rows {k0,k0+1}; lane>=16 -> N=lane-16, K={k0+2,k0+3}
//   C/D 16x16:  VGPR v -> M=v (lanes 0-15) / M=v+8 (lanes 16-31), N=lane&15
// ---------------------------------------------------------------------------
__global__ void sage_wmma_kernel(const float* __restrict__ agg,
                                 const float* __restrict__ inv,
                                 const float* __restrict__ hin,
                                 const float* __restrict__ Wl,
                                 const float* __restrict__ Wr,
                                 const float* __restrict__ bias,
                                 float* __restrict__ hout,
                                 int din, int dout, int nTilesN) {
  int wave  = (blockIdx.x * blockDim.x + threadIdx.x) >> 5;
  int lane  = threadIdx.x & 31;
  int mTile = wave / nTilesN;
  int nTile = wave % nTilesN;
  int mBase = mTile * 16;
  int nBase = nTile * 16;
  int half  = lane >> 4;     // 0: K pair {0,1}, 1: K pair {2,3}
  int l16   = lane & 15;

  int m = mBase + l16;       // A row handled by this lane
  int n = nBase + l16;       // B column handled by this lane
  float invm = inv[m];       // fused mean divide on the A operand
  const float* aggRow = agg + (size_t)m * din;
  const float* hinRow = hin + (size_t)m * din;

  v8f c = {0.f, 0.f, 0.f, 0.f, 0.f, 0.f, 0.f, 0.f};

  for (int k0 = 0; k0 < din; k0 += 4) {
    int kA = k0 + half * 2;
    v2f aL, aR, bL, bR;
    aL.x = aggRow[kA]     * invm;
    aL.y = aggRow[kA + 1] * invm;
    aR.x = hinRow[kA];
    aR.y = hinRow[kA + 1];
    bL.x = Wl[(size_t)kA       * dout + n];
    bL.y = Wl[(size_t)(kA + 1) * dout + n];
    bR.x = Wr[(size_t)kA       * dout + n];
    bR.y = Wr[(size_t)(kA + 1) * dout + n];
    // 8-arg form: (neg_a, A, neg_b, B, c_mod, C, reuse_a, reuse_b)
    c = __builtin_amdgcn_wmma_f32_16x16x4_f32(false, aL, false, bL,
                                              (short)0, c, false, false);
    c = __builtin_amdgcn_wmma_f32_16x16x4_f32(false, aR, false, bR,
                                              (short)0, c, false, false);
  }

  float bn = bias[n];
  #pragma unroll
  for (int v = 0; v < 8; ++v) {
    int mo = mBase + v + half * 8;
    float val = fmaxf(c[v] + bn, 0.0f);   // ReLU fused
    hout[(size_t)mo * dout + (nBase + l16)] = val;
  }
}

// ---------------------------------------------------------------------------
// Tiny classifier: out[n,:2] = h[n,:64] @ Wc + bc. Wc/bc staged in LDS.
// ---------------------------------------------------------------------------
__global__ void classifier_kernel(const float* __restrict__ h,
                                  const float* __restrict__ Wc,
                                  const float* __restrict__ bc,
                                  float* __restrict__ out, int n) {
  __shared__ float w[128];
  __shared__ float b[2];
  if (threadIdx.x < 128) w[threadIdx.x] = Wc[threadIdx.x];
  if (threadIdx.x < 2)   b[threadIdx.x] = bc[threadIdx.x];
  __syncthreads();
  int i = blockIdx.x * blockDim.x + threadIdx.x;
  if (i >= n) return;
  const float* hr = h + (size_t)i * 64;
  float o0 = b[0], o1 = b[1];
  #pragma unroll
  for (int k = 0; k < 64; ++k) {
    float hv = hr[k];
    o0 += hv * w[k * 2 + 0];
    o1 += hv * w[k * 2 + 1];
  }
  out[i * 2 + 0] = o0;
  out[i * 2 + 1] = o1;
}

// ---------------------------------------------------------------------------
// Orchestration. Input order (setup_inputs dict):
//  0:x 1:edge_index 2:Wl1 3:Wr1 4:b1 5:Wl2 6:Wr2 7:b2 8:Wl3 9:Wr3 10:b3 11:Wc 12:bc
// ---------------------------------------------------------------------------
extern "C" void kernel_launch(void* const* d_in, const int* in_sizes, int n_in,
                              void* d_out, int out_size, void* d_ws, size_t ws_size,
                              hipStream_t stream) {
  const float* x   = (const float*)d_in[0];
  const int*   ei  = (const int*)d_in[1];
  const int*   src = ei;             // edge_index[0]
  const int*   dst = ei + N_EDGES;   // edge_index[1]
  const float* Wl1 = (const float*)d_in[2];
  const float* Wr1 = (const float*)d_in[3];
  const float* b1  = (const float*)d_in[4];
  const float* Wl2 = (const float*)d_in[5];
  const float* Wr2 = (const float*)d_in[6];
  const float* b2  = (const float*)d_in[7];
  const float* Wl3 = (const float*)d_in[8];
  const float* Wr3 = (const float*)d_in[9];
  const float* b3  = (const float*)d_in[10];
  const float* Wc  = (const float*)d_in[11];
  const float* bc  = (const float*)d_in[12];
  float* out = (float*)d_out;

  // Workspace layout (floats): inv[N] | agg[N*128] | hA[N*128] | hB[N*128]
  float* inv = (float*)d_ws;
  float* agg = inv + N_NODES;
  float* hA  = agg + (size_t)N_NODES * HID;
  float* hB  = hA  + (size_t)N_NODES * HID;

  const int ZT = 256;

  // ---- degree -> inverse count (shared by all three layers) ----
  zero_f32<<<(N_NODES + ZT - 1) / ZT, ZT, 0, stream>>>(inv, N_NODES);
  degree_kernel<<<(N_EDGES + ZT - 1) / ZT, ZT, 0, stream>>>(dst, inv, N_EDGES);
  inv_kernel<<<(N_NODES + ZT - 1) / ZT, ZT, 0, stream>>>(inv, N_NODES);

  // scatter: one wave per edge, 8 waves per 256-thread block
  const int scatterBlocks = N_EDGES / 8;  // 800000 / 8 = 100000

  // ---- layer 1: x[.,96] -> hA[.,128] ----
  zero_f32<<<((N_NODES * IN_CH) + ZT - 1) / ZT, ZT, 0, stream>>>(agg, N_NODES * IN_CH);
  scatter_kernel<<<scatterBlocks, 256, 0, stream>>>(x, src, dst, agg, N_EDGES, IN_CH);
  {
    int nTilesN = HID / 16;                           // 8
    int waves   = (N_NODES / 16) * nTilesN;           // 25000
    sage_wmma_kernel<<<waves / 4, 128, 0, stream>>>(agg, inv, x, Wl1, Wr1, b1,
                                                    hA, IN_CH, HID, nTilesN);
  }

  // ---- layer 2: hA[.,128] -> hB[.,128] ----
  zero_f32<<<((N_NODES * HID) + ZT - 1) / ZT, ZT, 0, stream>>>(agg, N_NODES * HID);
  scatter_kernel<<<scatterBlocks, 256, 0, stream>>>(hA, src, dst, agg, N_EDGES, HID);
  {
    int nTilesN = HID / 16;                           // 8
    int waves   = (N_NODES / 16) * nTilesN;           // 25000
    sage_wmma_kernel<<<waves / 4, 128, 0, stream>>>(agg, inv, hA, Wl2, Wr2, b2,
                                                    hB, HID, HID, nTilesN);
  }

  // ---- layer 3: hB[.,128] -> hA[.,64] (reuse hA; h1 is dead) ----
  zero_f32<<<((N_NODES * HID) + ZT - 1) / ZT, ZT, 0, stream>>>(agg, N_NODES * HID);
  scatter_kernel<<<scatterBlocks, 256, 0, stream>>>(hB, src, dst, agg, N_EDGES, HID);
  {
    int nTilesN = (HID / 2) / 16;                     // 4
    int waves   = (N_NODES / 16) * nTilesN;           // 12500
    sage_wmma_kernel<<<waves / 4, 128, 0, stream>>>(agg, inv, hB, Wl3, Wr3, b3,
                                                    hA, HID, HID / 2, nTilesN);
  }

  // ---- classifier: hA[.,64] @ Wc + bc -> out[.,2] ----
  classifier_kernel<<<(N_NODES + ZT - 1) / ZT, ZT, 0, stream>>>(hA, Wc, bc, out, N_NODES);
}